// create_rnn_86019605004917
// MI455X (gfx1250) — compile-verified
//
#include <hip/hip_runtime.h>
#include <hip/hip_bf16.h>

// Problem constants (match reference): x[TS, BS, 1] -> H[TS, BS, 1]
#define TS   2048
#define BS   16384
#define TILE 32                 // timesteps staged per async LDS tile
#define NT   (TS / TILE)        // 64 tiles

// ---------------------------------------------------------------------------
// gfx1250 hardware tanh (TRANS unit). Guarded: builtin if present, else asm.
// ---------------------------------------------------------------------------
#if __has_builtin(__builtin_amdgcn_tanhf)
static __device__ __forceinline__ float fast_tanh(float x) {
  return __builtin_amdgcn_tanhf(x);
}
#else
static __device__ __forceinline__ float fast_tanh(float x) {
  float r;
  asm("v_tanh_f32 %0, %1" : "=v"(r) : "v"(x));
  return r;
}
#endif

// ---------------------------------------------------------------------------
// Async global->LDS copy (CDNA5, ASYNCcnt-tracked). 16B per lane.
// Builtin signature (from hipcc diagnostic): (as1 v4i*, as3 v4i*, imm, imm).
// Int-casts avoid illegal C++ addrspace conversions; low 32 bits of a flat
// shared pointer are the LDS byte offset (LDS aperture keeps offset in
// addr[31:0]).
// ---------------------------------------------------------------------------
typedef int v4i __attribute__((vector_size(16)));
typedef __attribute__((address_space(1))) v4i g_v4i;   // global
typedef __attribute__((address_space(3))) v4i l_v4i;   // LDS

#if __has_builtin(__builtin_amdgcn_global_load_async_to_lds_b128)
static __device__ __forceinline__ void async_copy_b128(const float* g, float* l) {
  __builtin_amdgcn_global_load_async_to_lds_b128(
      (g_v4i*)(unsigned long long)g,
      (l_v4i*)(unsigned)(unsigned long long)l,
      /*offset=*/0, /*cpol=*/0);
}
#else
static __device__ __forceinline__ void async_copy_b128(const float* g, float* l) {
  unsigned lds_off = (unsigned)(unsigned long long)l;
  asm volatile("global_load_async_to_lds_b128 %0, %1, off"
               :: "v"(lds_off), "v"(g)
               : "memory");
}
#endif

static __device__ __forceinline__ void wait_async0() {
#if __has_builtin(__builtin_amdgcn_s_wait_asynccnt)
  __builtin_amdgcn_s_wait_asynccnt(0);
  asm volatile("" ::: "memory");     // compiler barrier: LDS written behind its back
#else
  asm volatile("s_wait_asynccnt 0" ::: "memory");
#endif
}

// ---------------------------------------------------------------------------
// One wave (32 lanes) per workgroup; lane <-> one batch element's 2048-step
// sequential chain. Double-buffered async LDS staging of x tiles.
// ---------------------------------------------------------------------------
__global__ __launch_bounds__(32) void rnn_scan_kernel(
    const float* __restrict__ x,   // [TS*BS]
    const float* __restrict__ W1,  // [3]
    const float* __restrict__ b1,  // [3]
    const float* __restrict__ W2,  // [3]
    const float* __restrict__ b2,  // [3]
    const float* __restrict__ W3,  // [3]
    const float* __restrict__ b3,  // [1]
    float* __restrict__ out)       // [TS*BS]
{
  __shared__ float smem[2][TILE * 32];

  const int lane  = threadIdx.x;          // 0..31
  const int bbase = blockIdx.x * 32;      // batch base for this wave
  const int b     = bbase + lane;

  // Uniform weights -> SGPRs via s_load.
  const float a0 = W1[0], a1 = W1[1], a2 = W1[2];
  const float c0 = W2[0], c1 = W2[1], c2 = W2[2];
  const float d0 = b1[0] + b2[0], d1 = b1[1] + b2[1], d2 = b1[2] + b2[2];
  const float w0 = W3[0], w1 = W3[1], w2 = W3[2];
  const float bb = b3[0];

  // Per tile: 8 async b128 ops move 32 rows x 128B (each instr: 4 rows).
  const int sub  = lane & 7;        // 16B chunk within a 128B row
  const int rsel = lane >> 3;       // which of 4 rows this lane covers
  const int coff = sub << 2;        // float offset within row

  auto stage_tile = [&](int tt, int buf) {
#pragma unroll
    for (int k = 0; k < TILE / 4; ++k) {
      const int row = k * 4 + rsel;
      const float* g = x + (tt * TILE + row) * BS + bbase + coff;
      float* l = &smem[buf][row * 32 + coff];
      async_copy_b128(g, l);
    }
  };

  stage_tile(0, 0);

  float h = 0.0f;
  for (int tt = 0; tt < NT; ++tt) {
    const int buf = tt & 1;
    wait_async0();                       // current tile resident in LDS
    if (tt + 1 < NT) stage_tile(tt + 1, buf ^ 1);  // overlap next tile's DMA

    const float* __restrict__ xt = &smem[buf][0];
#pragma unroll
    for (int i = 0; i < TILE; ++i) {
      const float xv = xt[i * 32 + lane];            // ds_load, h-independent
      // h-independent parts (hoisted out of the dependent chain by unroll):
      const float e0 = fmaf(a0, xv, d0);
      const float e1 = fmaf(a1, xv, d1);
      const float e2 = fmaf(a2, xv, d2);
      // dependent chain: 1 fma -> v_tanh_f32 -> short tree reduce
      const float t0 = fast_tanh(fmaf(c0, h, e0));
      const float t1 = fast_tanh(fmaf(c1, h, e1));
      const float t2 = fast_tanh(fmaf(c2, h, e2));
      const float p0  = fmaf(w0, t0, bb);
      const float p12 = fmaf(w1, t1, w2 * t2);
      h = p0 + p12;
      out[(tt * TILE + i) * BS + b] = h;             // fire-and-forget store
    }
  }
}

extern "C" void kernel_launch(void* const* d_in, const int* in_sizes, int n_in,
                              void* d_out, int out_size, void* d_ws, size_t ws_size,
                              hipStream_t stream) {
  (void)in_sizes; (void)n_in; (void)out_size; (void)d_ws; (void)ws_size;
  const float* x  = (const float*)d_in[0];
  const float* W1 = (const float*)d_in[1];
  const float* b1 = (const float*)d_in[2];
  const float* W2 = (const float*)d_in[3];
  const float* b2 = (const float*)d_in[4];
  const float* W3 = (const float*)d_in[5];
  const float* b3 = (const float*)d_in[6];
  float* out = (float*)d_out;

  dim3 grid(BS / 32), block(32);
  rnn_scan_kernel<<<grid, block, 0, stream>>>(x, W1, b1, W2, b2, W3, b3, out);
}